// DYSON_1872605741758
// MI455X (gfx1250) — compile-verified
//
#include <hip/hip_runtime.h>
#include <hip/hip_bf16.h>
#include <math.h>

typedef float v2f __attribute__((ext_vector_type(2)));
typedef float v8f __attribute__((ext_vector_type(8)));

#define Bn   512
#define Cn   1000
#define Dn   512
#define Kn   10
#define CPAD 1008   // 63 tiles of 16

// ---------------------------------------------------------------------------
// Kernel 1: per-class prep.
//   t[c,d]   = sqrt(N_c*ex2^2 - ex1^2)
//   rd[c,d]  = exp(min_d t - t) / sum_d exp(min_d t - t)   (== softmax(max-t))
//   WP[c,d]  = -2 * rd[c,d] * protos[c,d]
//   AB[c]    = sum_d rd[c,d] * protos[c,d]^2
// One block (256 threads) per class; each thread owns 2 of the 512 features.
// ---------------------------------------------------------------------------
__global__ void __launch_bounds__(256) dyson_prep_kernel(
    const float* __restrict__ protos,
    const float* __restrict__ ex2,
    const float* __restrict__ ex1,
    const int*   __restrict__ cls_num,
    float* __restrict__ RD,
    float* __restrict__ WP,
    float* __restrict__ AB) {
  __shared__ float red[256];
  const int c   = blockIdx.x;
  const int tid = threadIdx.x;
  const int d0  = tid;
  const int d1  = tid + 256;
  const float N = (float)cls_num[c];

  float e2 = ex2[c * Dn + d0], e1 = ex1[c * Dn + d0];
  float t0 = sqrtf(N * e2 * e2 - e1 * e1);
  e2 = ex2[c * Dn + d1]; e1 = ex1[c * Dn + d1];
  float t1 = sqrtf(N * e2 * e2 - e1 * e1);

  // block-wide min of t
  red[tid] = fminf(t0, t1);
  __syncthreads();
  for (int s = 128; s > 0; s >>= 1) {
    if (tid < s) red[tid] = fminf(red[tid], red[tid + s]);
    __syncthreads();
  }
  const float tmin = red[0];
  __syncthreads();

  const float w0 = expf(tmin - t0);
  const float w1 = expf(tmin - t1);

  // block-wide sum of exp
  red[tid] = w0 + w1;
  __syncthreads();
  for (int s = 128; s > 0; s >>= 1) {
    if (tid < s) red[tid] += red[tid + s];
    __syncthreads();
  }
  const float inv_sum = 1.0f / red[0];
  __syncthreads();

  const float rd0 = w0 * inv_sum;
  const float rd1 = w1 * inv_sum;
  const float p0  = protos[c * Dn + d0];
  const float p1  = protos[c * Dn + d1];

  RD[c * Dn + d0] = rd0;
  RD[c * Dn + d1] = rd1;
  WP[c * Dn + d0] = -2.0f * rd0 * p0;
  WP[c * Dn + d1] = -2.0f * rd1 * p1;

  // block-wide sum of rd * p^2 -> bias
  red[tid] = rd0 * p0 * p0 + rd1 * p1 * p1;
  __syncthreads();
  for (int s = 128; s > 0; s >>= 1) {
    if (tid < s) red[tid] += red[tid + s];
    __syncthreads();
  }
  if (tid == 0) AB[c] = red[0];
}

// ---------------------------------------------------------------------------
// Kernel 2: fused double GEMM via fp32 WMMA.
//   SIMI[b,c] = AB[c] + sum_d x^2[b,d]*RD[c,d] + sum_d x[b,d]*WP[c,d]
// 8 waves per block, one 16x16 output tile per wave.
// 2016 tiles = 32 (M) x 63 (N) = 252 blocks x 8 waves exactly.
// A layout (16x4 fp32): lane holds row M=lane&15, VGPR pair holds K = 2*(lane>>4)+{0,1}
// B layout (4x16 fp32): lane holds col N=lane&15, same K striping.
// C/D layout: VGPR v, lane L -> (M = v + 8*(L>>4), N = L&15).
// ---------------------------------------------------------------------------
__global__ void __launch_bounds__(256) dyson_gemm_wmma_kernel(
    const float* __restrict__ x,
    const float* __restrict__ RD,
    const float* __restrict__ WP,
    const float* __restrict__ AB,
    float* __restrict__ SIMI) {
  const int lane  = threadIdx.x & 31;
  const int wave  = threadIdx.x >> 5;
  const int tile  = blockIdx.x * 8 + wave;
  const int mtile = tile & 31;   // 0..31
  const int ntile = tile >> 5;   // 0..62
  const int mbase = mtile << 4;
  const int cbase = ntile << 4;
  const int half  = lane >> 4;
  const int l16   = lane & 15;

  const int arow   = mbase + l16;
  const int bcol   = cbase + l16;
  const int bclamp = (bcol < Cn) ? bcol : (Cn - 1);

  const float* xp  = x  + (size_t)arow   * Dn + 2 * half;
  const float* rdp = RD + (size_t)bclamp * Dn + 2 * half;
  const float* wpp = WP + (size_t)bclamp * Dn + 2 * half;

  const float bias = AB[bclamp];
  v8f acc;
#pragma unroll
  for (int i = 0; i < 8; ++i) acc[i] = bias;

  for (int k = 0; k < Dn; k += 4) {
    __builtin_prefetch(xp + k + 64, 0, 1);   // global_prefetch of upcoming A rows
    v2f xa  = *(const v2f*)(xp  + k);
    v2f brd = *(const v2f*)(rdp + k);
    v2f bwp = *(const v2f*)(wpp + k);
    v2f xs  = xa * xa;
    // D = x^2 * rd^T + C
    acc = __builtin_amdgcn_wmma_f32_16x16x4_f32(
        /*neg_a=*/false, xs, /*neg_b=*/false, brd,
        /*c_mod=*/(short)0, acc, /*reuse_a=*/false, /*reuse_b=*/false);
    // D += x * (-2 rd p)^T
    acc = __builtin_amdgcn_wmma_f32_16x16x4_f32(
        /*neg_a=*/false, xa, /*neg_b=*/false, bwp,
        /*c_mod=*/(short)0, acc, /*reuse_a=*/false, /*reuse_b=*/false);
  }

#pragma unroll
  for (int v = 0; v < 8; ++v) {
    const int row = mbase + v + 8 * half;
    SIMI[(size_t)row * CPAD + bcol] = acc[v];
  }
}

// ---------------------------------------------------------------------------
// Kernel 3: per-row 10-smallest (stable: ties keep lower class index, matching
// jax.lax.top_k on -simi), conf = sum/vals, argmax (first max) -> label.
// Output: [0,512) predict (as float), [512, 512+512*10) topk_conf row-major.
// ---------------------------------------------------------------------------
__global__ void __launch_bounds__(256) dyson_topk_kernel(
    const float* __restrict__ SIMI,
    const int*   __restrict__ proto_label,
    float* __restrict__ out) {
  const int b = blockIdx.x * blockDim.x + threadIdx.x;
  if (b >= Bn) return;

  float vals[Kn];
  int   idxs[Kn];
#pragma unroll
  for (int i = 0; i < Kn; ++i) { vals[i] = 3.402823466e38f; idxs[i] = 0; }

  const float* row = SIMI + (size_t)b * CPAD;
  for (int c = 0; c < Cn; ++c) {
    const float s = row[c];
    if (s < vals[Kn - 1]) {
      int j = Kn - 1;
      while (j > 0 && s < vals[j - 1]) {
        vals[j] = vals[j - 1];
        idxs[j] = idxs[j - 1];
        --j;
      }
      vals[j] = s;
      idxs[j] = c;
    }
  }

  float S = 0.0f;
#pragma unroll
  for (int i = 0; i < Kn; ++i) S += vals[i];

  float conf[Kn];
#pragma unroll
  for (int i = 0; i < Kn; ++i) conf[i] = S / vals[i];

  int best = 0;
  float bc = conf[0];
#pragma unroll
  for (int i = 1; i < Kn; ++i)
    if (conf[i] > bc) { bc = conf[i]; best = i; }

  out[b] = (float)proto_label[idxs[best]];
#pragma unroll
  for (int i = 0; i < Kn; ++i) out[Bn + b * Kn + i] = conf[i];
}

// ---------------------------------------------------------------------------
extern "C" void kernel_launch(void* const* d_in, const int* in_sizes, int n_in,
                              void* d_out, int out_size, void* d_ws, size_t ws_size,
                              hipStream_t stream) {
  const float* x           = (const float*)d_in[0];
  const float* protos      = (const float*)d_in[1];
  const float* ex2         = (const float*)d_in[2];
  const float* ex1         = (const float*)d_in[3];
  const int*   cls_num     = (const int*)d_in[4];
  const int*   proto_label = (const int*)d_in[5];

  float* ws   = (float*)d_ws;
  float* RD   = ws;                          // [C, D]    = 512000 floats
  float* WP   = RD + (size_t)Cn * Dn;        // [C, D]    = 512000 floats
  float* AB   = WP + (size_t)Cn * Dn;        // [C] (pad to 1024)
  float* SIMI = AB + 1024;                   // [B, CPAD] = 516096 floats

  dyson_prep_kernel<<<Cn, 256, 0, stream>>>(protos, ex2, ex1, cls_num, RD, WP, AB);
  dyson_gemm_wmma_kernel<<<252, 256, 0, stream>>>(x, RD, WP, AB, SIMI);
  dyson_topk_kernel<<<(Bn + 255) / 256, 256, 0, stream>>>(SIMI, proto_label, (float*)d_out);
}